// PositionNet_61529701482960
// MI455X (gfx1250) — compile-verified
//
#include <hip/hip_runtime.h>
#include <math.h>

typedef __attribute__((ext_vector_type(2))) float    v2f;
typedef __attribute__((ext_vector_type(8))) float    v8f;
typedef __attribute__((ext_vector_type(2))) unsigned v2u;

#define BN      4
#define NBOX    30
#define IN_DIM  768
#define POS_DIM 64
#define XDIM    (IN_DIM + POS_DIM)   // 832
#define H_DIM   512
#define OUT_DIM 768
#define MROWS   (BN * NBOX)          // 120
#define MPAD    128                  // 8 tiles of 16

// ---------------------------------------------------------------------------
// 1) Per-pixel coverage bitmask: bit n (n < NBOX-1) set if valid box n covers
//    the pixel; bit NBOX-1 = "null slot" set iff no box covers the pixel.
// ---------------------------------------------------------------------------
__global__ void coverage_masks_kernel(const float* __restrict__ boxes,
                                      const float* __restrict__ masks,
                                      unsigned* __restrict__ out, int size) {
  __shared__ float sb[NBOX * 4];
  __shared__ float sm[NBOX];
  const int b   = blockIdx.y;
  const int tid = threadIdx.x;
  if (tid < NBOX * 4) sb[tid] = boxes[b * NBOX * 4 + tid];
  if (tid < NBOX)     sm[tid] = masks[b * NBOX + tid];
  __syncthreads();

  const int HW  = size * size;
  const int pix = blockIdx.x * blockDim.x + tid;
  if (pix >= HW) return;
  const float fy = (float)(pix / size);
  const float fx = (float)(pix % size);
  const float fs = (float)size;

  unsigned m = 0u;
#pragma unroll 1
  for (int n = 0; n < NBOX - 1; ++n) {
    if (sm[n] == 1.0f) {
      // jnp.round = round-half-to-even -> rintf under default RNE mode
      float x1 = rintf(sb[n * 4 + 0] * fs);
      float y1 = rintf(sb[n * 4 + 1] * fs);
      float x2 = rintf(sb[n * 4 + 2] * fs);
      float y2 = rintf(sb[n * 4 + 3] * fs);
      if (fy >= y1 && fy < y2 && fx >= x1 && fx < x2) m |= (1u << n);
    }
  }
  if (m == 0u) m = (1u << (NBOX - 1));   // null slot covers uncovered pixels
  out[b * HW + pix] = m;
}

// ---------------------------------------------------------------------------
// 2) Fill full sim matrix [B, HW+N, HW+N]:
//    [:HW,:HW]  = (mask_p & mask_q) ? 1 : 0   (== min(am^T am, 1))
//    [:HW,HW:]  = am^T  (bit n of mask_p, null bit included)
//    [HW:, :]   = 0
//    Store-bandwidth bound (~290 MB): b64 stores (S is even -> every row base
//    is 8B-aligned; S % 4 == 2 forbids b128) with NT hint so the 272 MB sim64
//    streams past L2 while the 16 KB/batch mask array stays cache-resident.
// ---------------------------------------------------------------------------
__global__ void fill_sim_kernel(const unsigned* __restrict__ masks,
                                float* __restrict__ out, int size) {
  const int HW = size * size;
  const int S  = HW + NBOX;            // even
  const int b  = blockIdx.y;
  const int r  = blockIdx.x;
  float* row = out + ((size_t)b * S + r) * S;
  v2f*   row2 = (v2f*)row;             // 8B-aligned

  if (r >= HW) {                       // bottom N rows are all zeros
    const v2f z = {0.0f, 0.0f};
    for (int q2 = threadIdx.x; q2 < (S >> 1); q2 += blockDim.x)
      __builtin_nontemporal_store(z, row2 + q2);
    return;
  }
  const unsigned* mb  = masks + b * HW;
  const v2u*      mb2 = (const v2u*)mb;
  const unsigned  mp  = mb[r];
  for (int q2 = threadIdx.x; q2 < (HW >> 1); q2 += blockDim.x) {
    const v2u mq = mb2[q2];
    v2f v;
    v.x = (mp & mq.x) ? 1.0f : 0.0f;
    v.y = (mp & mq.y) ? 1.0f : 0.0f;
    __builtin_nontemporal_store(v, row2 + q2);
  }
  for (int n = threadIdx.x; n < NBOX; n += blockDim.x)
    row[HW + n] = (float)((mp >> n) & 1u);
}

// ---------------------------------------------------------------------------
// 3) Build MLP input X [MPAD x 832]: masked (pe + pos_table) || masked Fourier.
//    Rows 120..127 zero-padded (row-independent GEMM -> no contamination).
// ---------------------------------------------------------------------------
__global__ void build_x_kernel(const float* __restrict__ boxes,
                               const float* __restrict__ masks,
                               const float* __restrict__ pos_emb,
                               const float* __restrict__ pos_table,
                               const float* __restrict__ null_positive,
                               const float* __restrict__ null_position,
                               float* __restrict__ X) {
  const int idx = blockIdx.x * blockDim.x + threadIdx.x;
  if (idx >= MPAD * XDIM) return;
  const int row = idx / XDIM, c = idx % XDIM;
  if (row >= MROWS) { X[idx] = 0.0f; return; }
  const int b = row / NBOX, n = row % NBOX;
  const float m = masks[b * NBOX + n];
  float val;
  if (c < IN_DIM) {
    float v = pos_emb[(size_t)(b * NBOX + n) * IN_DIM + c] + pos_table[(size_t)n * IN_DIM + c];
    val = m * v + (1.0f - m) * null_positive[c];
  } else {
    const int j = c - IN_DIM;            // 0..63 : layout f*8 + k
    const int f = j >> 3, k = j & 7;     // k<4 -> sin(freq*box[k]), k>=4 -> cos(freq*box[k-4])
    const float freq = powf(100.0f, (float)f * 0.125f);
    const float bx   = boxes[(size_t)(b * NBOX + n) * 4 + (k & 3)];
    const float t    = freq * bx;
    const float s    = (k < 4) ? sinf(t) : cosf(t);
    val = m * s + (1.0f - m) * null_position[j];
  }
  X[idx] = val;
}

// ---------------------------------------------------------------------------
// 4) fp32 WMMA GEMM: Out[M x Nc] = silu?(A[MPAD x K] @ W[K x Nc] + bias)
//    One wave per 16x16 tile, V_WMMA_F32_16X16X4_F32 over K/4 steps.
//    A frag: lane(l): row l%16, K = k0 + (l/16)*2 + {0,1}   (ISA 7.12.2)
//    C/D   : vgpr v -> row v + (l/16)*8, col l%16
// ---------------------------------------------------------------------------
__global__ void gemm_wmma_kernel(const float* __restrict__ A,
                                 const float* __restrict__ W,
                                 const float* __restrict__ bias,
                                 float* __restrict__ Out,
                                 int K, int Ncols, int m_valid, int apply_silu) {
  const int wave = threadIdx.x >> 5;            // 8 waves / block
  const int lane = threadIdx.x & 31;
  const int tiles_n = Ncols >> 4;
  const int tile = blockIdx.x * 8 + wave;       // grid sized exactly
  const int tm = tile / tiles_n;
  const int tn = tile % tiles_n;
  const int half = lane >> 4;                   // 0 or 1
  const int r    = lane & 15;

  v8f acc = {0.f, 0.f, 0.f, 0.f, 0.f, 0.f, 0.f, 0.f};
  const float* arow = A + (size_t)(tm * 16 + r) * K + half * 2;
  const float* wcol = W + (size_t)(half * 2) * Ncols + tn * 16 + r;

  for (int k0 = 0; k0 < K; k0 += 4) {
    v2f a;
    a.x = arow[k0];
    a.y = arow[k0 + 1];
    v2f bf;
    bf.x = wcol[(size_t)k0 * Ncols];
    bf.y = wcol[(size_t)(k0 + 1) * Ncols];
    acc = __builtin_amdgcn_wmma_f32_16x16x4_f32(false, a, false, bf,
                                                (short)0, acc, false, false);
  }

  const int col = tn * 16 + r;
  const float bcol = bias[col];
#pragma unroll
  for (int v = 0; v < 8; ++v) {
    const int row = tm * 16 + v + half * 8;
    float x = acc[v] + bcol;
    if (apply_silu) x = x / (1.0f + expf(-x));  // silu = x * sigmoid(x)
    if (row < m_valid) Out[(size_t)row * Ncols + col] = x;
  }
}

// ---------------------------------------------------------------------------
extern "C" void kernel_launch(void* const* d_in, const int* in_sizes, int n_in,
                              void* d_out, int out_size, void* d_ws, size_t ws_size,
                              hipStream_t stream) {
  const float* boxes         = (const float*)d_in[0];
  const float* masksf        = (const float*)d_in[1];
  const float* pos_emb       = (const float*)d_in[2];
  const float* pos_table     = (const float*)d_in[3];
  const float* W1            = (const float*)d_in[4];
  const float* b1            = (const float*)d_in[5];
  const float* W2            = (const float*)d_in[6];
  const float* b2            = (const float*)d_in[7];
  const float* W3            = (const float*)d_in[8];
  const float* b3            = (const float*)d_in[9];
  const float* null_positive = (const float*)d_in[10];
  const float* null_position = (const float*)d_in[11];

  const int HW64 = 64 * 64, HW32 = 32 * 32;
  const int S64 = HW64 + NBOX;          // 4126
  const int S32 = HW32 + NBOX;          // 1054

  float* out   = (float*)d_out;
  float* objs  = out;                                        // [4,30,768]
  float* sim64 = out + (size_t)BN * NBOX * OUT_DIM;          // [4,4126,4126]
  float* sim32 = sim64 + (size_t)BN * S64 * S64;             // [4,1054,1054]

  char* ws = (char*)d_ws;
  float*    X    = (float*)ws;  ws += sizeof(float) * (size_t)MPAD * XDIM;
  float*    H1   = (float*)ws;  ws += sizeof(float) * (size_t)MPAD * H_DIM;
  float*    H2   = (float*)ws;  ws += sizeof(float) * (size_t)MPAD * H_DIM;
  unsigned* cm64 = (unsigned*)ws; ws += sizeof(unsigned) * (size_t)BN * HW64;
  unsigned* cm32 = (unsigned*)ws;

  // --- sim masks (store-bandwidth bound; the roofline floor) ---
  {
    dim3 g((HW64 + 255) / 256, BN);
    coverage_masks_kernel<<<g, 256, 0, stream>>>(boxes, masksf, cm64, 64);
  }
  {
    dim3 g((HW32 + 255) / 256, BN);
    coverage_masks_kernel<<<g, 256, 0, stream>>>(boxes, masksf, cm32, 32);
  }
  {
    dim3 g(S64, BN);
    fill_sim_kernel<<<g, 256, 0, stream>>>(cm64, sim64, 64);
  }
  {
    dim3 g(S32, BN);
    fill_sim_kernel<<<g, 256, 0, stream>>>(cm32, sim32, 32);
  }

  // --- MLP with fp32 WMMA ---
  {
    const int n = MPAD * XDIM;
    build_x_kernel<<<(n + 255) / 256, 256, 0, stream>>>(
        boxes, masksf, pos_emb, pos_table, null_positive, null_position, X);
  }
  // layer 1: [128x832]@[832x512]  -> 8*32 = 256 tiles -> 32 blocks
  gemm_wmma_kernel<<<32, 256, 0, stream>>>(X, W1, b1, H1, XDIM, H_DIM, MPAD, 1);
  // layer 2: [128x512]@[512x512]  -> 256 tiles -> 32 blocks
  gemm_wmma_kernel<<<32, 256, 0, stream>>>(H1, W2, b2, H2, H_DIM, H_DIM, MPAD, 1);
  // layer 3: [128x512]@[512x768]  -> 8*48 = 384 tiles -> 48 blocks (store rows<120)
  gemm_wmma_kernel<<<48, 256, 0, stream>>>(H2, W3, b3, objs, H_DIM, OUT_DIM, MROWS, 0);
}